// Seq2Seq_42460046688258
// MI455X (gfx1250) — compile-verified
//
#include <hip/hip_runtime.h>
#include <cmath>

// ---------------- problem constants ----------------
#define V_SZ 50257
#define I_SZ 1024
#define N_SZ 1024
#define B_SZ 64
#define T_SZ 2048
#define G3N  (3 * N_SZ)
#define NCHUNK 16
#define TCHUNK (T_SZ / NCHUNK)   // 128

typedef float v2f __attribute__((ext_vector_type(2)));
typedef float v8f __attribute__((ext_vector_type(8)));

// ---------------------------------------------------------------------------
// Kernel 1: embedding gather + concat(word_embedded, last_context) -> rnn_input
// ---------------------------------------------------------------------------
__global__ void k_embed_concat(const int* __restrict__ word,
                               const float* __restrict__ last_ctx,
                               const float* __restrict__ emb,
                               float* __restrict__ rnn) {
    int b = blockIdx.x;
    int w = word[b];
    for (int i = threadIdx.x; i < I_SZ; i += blockDim.x) {
        rnn[b * 2 * N_SZ + i]        = emb[(size_t)w * I_SZ + i];
        rnn[b * 2 * N_SZ + N_SZ + i] = last_ctx[b * N_SZ + i];
    }
}

// ---------------------------------------------------------------------------
// Kernel 2: fp32 WMMA GEMM  out[64 x Nout] = A[64 x K] @ W[Nout x K]^T + bias
// V_WMMA_F32_16X16X4_F32.  One wave owns one 16-wide column tile and all four
// 16-row M tiles; each B fragment feeds 4 WMMAs.  The k-loop is software-
// pipelined with a 1-deep register double buffer so next-step fragment loads
// overlap the current WMMA group (avoids the load->s_wait_loadcnt0->wmma
// lockstep the naive loop produced).  Each lane streams one W row
// sequentially -> full cacheline use; W is read from HBM exactly once.
// ---------------------------------------------------------------------------
__global__ __launch_bounds__(256) void k_wmma_gemm_awt(
    const float* __restrict__ A, const float* __restrict__ W,
    const float* __restrict__ bias, float* __restrict__ out,
    int K, int Nout) {
    const int lane = threadIdx.x & 31;
    const int wid  = threadIdx.x >> 5;
    const int half = lane >> 4;      // 0: K=0,1  1: K=2,3 (A/B frag halves)
    const int lm   = lane & 15;

    const int ntile = blockIdx.x * 8 + wid;
    const int n0 = ntile * 16;
    if (n0 >= Nout) return;          // uniform per wave -> EXEC stays all-1s

    const int ncol = n0 + lm;
    const int ncl  = (ncol < Nout) ? ncol : (Nout - 1);  // clamp for loads

    v8f acc[4];
    const v8f z = {0.f, 0.f, 0.f, 0.f, 0.f, 0.f, 0.f, 0.f};
#pragma unroll
    for (int mt = 0; mt < 4; ++mt) acc[mt] = z;

    // per-lane stream bases (offset by this lane-half's K pair)
    const float* wp = W + (size_t)ncl * K + 2 * half;
    const float* ap[4];
#pragma unroll
    for (int mt = 0; mt < 4; ++mt)
        ap[mt] = A + (size_t)(mt * 16 + lm) * K + 2 * half;

    // prologue: fragments for k0 = 0
    v2f bcur = *(const v2f*)wp;
    v2f acur[4];
#pragma unroll
    for (int mt = 0; mt < 4; ++mt) acur[mt] = *(const v2f*)ap[mt];

    // steady state: issue k0+4 loads, then consume k0 fragments
    for (int k0 = 0; k0 < K - 4; k0 += 4) {
        v2f bnxt = *(const v2f*)(wp + k0 + 4);
        v2f anxt[4];
#pragma unroll
        for (int mt = 0; mt < 4; ++mt)
            anxt[mt] = *(const v2f*)(ap[mt] + k0 + 4);
#pragma unroll
        for (int mt = 0; mt < 4; ++mt)
            acc[mt] = __builtin_amdgcn_wmma_f32_16x16x4_f32(
                false, acur[mt], false, bcur, (short)0, acc[mt], false, false);
        bcur = bnxt;
#pragma unroll
        for (int mt = 0; mt < 4; ++mt) acur[mt] = anxt[mt];
    }
    // epilogue: last k-step
#pragma unroll
    for (int mt = 0; mt < 4; ++mt)
        acc[mt] = __builtin_amdgcn_wmma_f32_16x16x4_f32(
            false, acur[mt], false, bcur, (short)0, acc[mt], false, false);

    const float bv = bias[ncl];
    if (ncol < Nout) {
#pragma unroll
        for (int mt = 0; mt < 4; ++mt)
#pragma unroll
            for (int j = 0; j < 8; ++j) {
                int row = mt * 16 + 8 * half + j;   // C/D layout: M = v + 8*half
                out[(size_t)row * Nout + ncol] = acc[mt][j] + bv;
            }
    }
}

// ---------------------------------------------------------------------------
// Kernel 3: GRU gate nonlinearity -> h_new (to ws, to cat[:, :N], to d_out)
// ---------------------------------------------------------------------------
__global__ void k_gru_gates(const float* __restrict__ gx, const float* __restrict__ gh,
                            const float* __restrict__ h_prev,
                            float* __restrict__ hnew, float* __restrict__ cat,
                            float* __restrict__ hout) {
    int idx = blockIdx.x * blockDim.x + threadIdx.x;   // B*N
    int b = idx >> 10, n = idx & (N_SZ - 1);
    float xr = gx[b * G3N + n];
    float xz = gx[b * G3N + N_SZ + n];
    float xn = gx[b * G3N + 2 * N_SZ + n];
    float hr = gh[b * G3N + n];
    float hz = gh[b * G3N + N_SZ + n];
    float hn = gh[b * G3N + 2 * N_SZ + n];
    float r = 1.f / (1.f + expf(-(xr + hr)));
    float z = 1.f / (1.f + expf(-(xz + hz)));
    float nn = tanhf(xn + r * hn);
    float h = (1.f - z) * nn + z * h_prev[idx];
    hnew[idx] = h;
    cat[b * 2 * N_SZ + n] = h;
    hout[idx] = h;
}

// ---------------------------------------------------------------------------
// Kernel 4: attention phase A — one pass over enc per 128-t chunk.
// Per block (chunk, b): dot products (wave-parallel), chunk-local softmax
// numerators p_t (written to attn out, rescaled later), local max/sum, and
// partial context  pctx[c,b,:] = sum_{t in c} p_t * enc[t,b,:].
// enc slice (512KB) is touched twice but the 2nd read hits L2.
// ---------------------------------------------------------------------------
__global__ __launch_bounds__(256) void k_attn_partial(
    const float* __restrict__ hnew, const float* __restrict__ enc,
    float* __restrict__ attn_w, float* __restrict__ pctx,
    float* __restrict__ lmax, float* __restrict__ lsum) {
    __shared__ float s_p[TCHUNK];
    __shared__ float red[TCHUNK];
    const int chunk = blockIdx.x, b = blockIdx.y;
    const int tid = threadIdx.x;
    const int lane = tid & 31, wid = tid >> 5;
    const float4* hb = (const float4*)(hnew + b * N_SZ);

    // --- dots: each wave computes 16 of the 128 t's in this chunk ---
    for (int j = 0; j < TCHUNK / 8; ++j) {
        int lt = wid * (TCHUNK / 8) + j;
        int t = chunk * TCHUNK + lt;
        const float4* eb = (const float4*)(enc + ((size_t)t * B_SZ + b) * N_SZ);
        float acc = 0.f;
        for (int i = lane; i < N_SZ / 4; i += 32) {   // 8 iters, coalesced
            float4 hv = hb[i], ev = eb[i];
            acc += hv.x * ev.x + hv.y * ev.y + hv.z * ev.z + hv.w * ev.w;
        }
        for (int off = 16; off > 0; off >>= 1) acc += __shfl_xor(acc, off, 32);
        if (lane == 0) s_p[lt] = acc;
    }
    __syncthreads();

    // --- chunk-local max ---
    if (tid < TCHUNK) red[tid] = s_p[tid];
    __syncthreads();
    for (int st = TCHUNK / 2; st > 0; st >>= 1) {
        if (tid < st) red[tid] = fmaxf(red[tid], red[tid + st]);
        __syncthreads();
    }
    const float m = red[0];
    __syncthreads();

    // --- p_t = exp(s - m), store, and chunk-local sum ---
    if (tid < TCHUNK) {
        float p = expf(s_p[tid] - m);
        s_p[tid] = p;
        attn_w[(size_t)b * T_SZ + chunk * TCHUNK + tid] = p;  // rescaled in phase B
        red[tid] = p;
    }
    __syncthreads();
    for (int st = TCHUNK / 2; st > 0; st >>= 1) {
        if (tid < st) red[tid] += red[tid + st];
        __syncthreads();
    }
    if (tid == 0) {
        lmax[b * NCHUNK + chunk] = m;
        lsum[b * NCHUNK + chunk] = red[0];
    }
    __syncthreads();

    // --- partial context: 256 threads x float4 = 1024 n's ---
    float4 acc = make_float4(0.f, 0.f, 0.f, 0.f);
    for (int lt = 0; lt < TCHUNK; ++lt) {
        float p = s_p[lt];
        const float4 e = ((const float4*)(enc +
            ((size_t)(chunk * TCHUNK + lt) * B_SZ + b) * N_SZ))[tid];
        acc.x += p * e.x; acc.y += p * e.y; acc.z += p * e.z; acc.w += p * e.w;
    }
    ((float4*)(pctx + ((size_t)chunk * B_SZ + b) * N_SZ))[tid] = acc;
}

// ---------------------------------------------------------------------------
// Kernel 5: attention phase B — combine 16 chunk partials (online softmax),
// rescale attn weights in place, emit context (to d_out and cat[:, N:2N]).
// ---------------------------------------------------------------------------
__global__ void k_attn_combine(const float* __restrict__ lmax, const float* __restrict__ lsum,
                               const float* __restrict__ pctx,
                               float* __restrict__ attn_w, float* __restrict__ ctx_out,
                               float* __restrict__ cat) {
    __shared__ float scale[NCHUNK];
    const int b = blockIdx.x, tid = threadIdx.x;
    if (tid == 0) {
        float gm = -3.402823466e38f;
        for (int c = 0; c < NCHUNK; ++c) gm = fmaxf(gm, lmax[b * NCHUNK + c]);
        float gs = 0.f;
        for (int c = 0; c < NCHUNK; ++c)
            gs += lsum[b * NCHUNK + c] * expf(lmax[b * NCHUNK + c] - gm);
        float inv = 1.f / gs;
        for (int c = 0; c < NCHUNK; ++c)
            scale[c] = expf(lmax[b * NCHUNK + c] - gm) * inv;
    }
    __syncthreads();
    for (int t = tid; t < T_SZ; t += 256)
        attn_w[(size_t)b * T_SZ + t] *= scale[t / TCHUNK];
    // context: N/4 = 256 float4's
    {
        float4 acc = make_float4(0.f, 0.f, 0.f, 0.f);
        for (int c = 0; c < NCHUNK; ++c) {
            float s = scale[c];
            float4 p = ((const float4*)(pctx + ((size_t)c * B_SZ + b) * N_SZ))[tid];
            acc.x += s * p.x; acc.y += s * p.y; acc.z += s * p.z; acc.w += s * p.w;
        }
        ((float4*)(ctx_out + (size_t)b * N_SZ))[tid] = acc;
        ((float4*)(cat + (size_t)b * 2 * N_SZ + N_SZ))[tid] = acc;
    }
}

// ---------------------------------------------------------------------------
// Kernel 6: row-wise log-softmax over V=50257
// ---------------------------------------------------------------------------
__global__ void k_log_softmax(const float* __restrict__ logits, float* __restrict__ out) {
    __shared__ float red[256];
    const int b = blockIdx.x, tid = threadIdx.x;
    const float* row = logits + (size_t)b * V_SZ;
    float m = -3.402823466e38f;
    for (int v = tid; v < V_SZ; v += 256) m = fmaxf(m, row[v]);
    red[tid] = m; __syncthreads();
    for (int st = 128; st > 0; st >>= 1) {
        if (tid < st) red[tid] = fmaxf(red[tid], red[tid + st]);
        __syncthreads();
    }
    m = red[0]; __syncthreads();
    float s = 0.f;
    for (int v = tid; v < V_SZ; v += 256) s += expf(row[v] - m);
    red[tid] = s; __syncthreads();
    for (int st = 128; st > 0; st >>= 1) {
        if (tid < st) red[tid] += red[tid + st];
        __syncthreads();
    }
    const float logZ = m + logf(red[0]);
    for (int v = tid; v < V_SZ; v += 256)
        out[(size_t)b * V_SZ + v] = row[v] - logZ;
}

// ---------------------------------------------------------------------------
extern "C" void kernel_launch(void* const* d_in, const int* in_sizes, int n_in,
                              void* d_out, int out_size, void* d_ws, size_t ws_size,
                              hipStream_t stream) {
    (void)in_sizes; (void)n_in; (void)out_size; (void)ws_size;
    const int*   word     = (const int*)d_in[0];
    const float* last_ctx = (const float*)d_in[1];
    const float* last_h   = (const float*)d_in[2];   // (1,B,N) -> B x N
    const float* enc      = (const float*)d_in[3];   // (T,B,N)
    const float* emb      = (const float*)d_in[4];
    const float* W_ih     = (const float*)d_in[5];   // (3N, 2N)
    const float* W_hh     = (const float*)d_in[6];   // (3N, N)
    const float* b_ih     = (const float*)d_in[7];
    const float* b_hh     = (const float*)d_in[8];
    const float* W_out    = (const float*)d_in[9];   // (V, 2N)
    const float* b_out    = (const float*)d_in[10];

    float* out = (float*)d_out;
    float* ws  = (float*)d_ws;

    // workspace layout (floats), ~20 MB total
    float* rnn    = ws;                               // B x 2N
    float* gx     = rnn    + B_SZ * 2 * N_SZ;         // B x 3N
    float* gh     = gx     + B_SZ * G3N;              // B x 3N
    float* hnew   = gh     + B_SZ * G3N;              // B x N
    float* cat    = hnew   + B_SZ * N_SZ;             // B x 2N  [h_new | context]
    float* pctx   = cat    + B_SZ * 2 * N_SZ;         // NCHUNK x B x N
    float* lmax   = pctx   + (size_t)NCHUNK * B_SZ * N_SZ;
    float* lsum   = lmax   + NCHUNK * B_SZ;
    float* logits = lsum   + NCHUNK * B_SZ;           // B x V

    // output layout: [log_probs BxV | context BxN | h_new BxN | attn BxT]
    float* out_logp = out;
    float* out_ctx  = out_logp + (size_t)B_SZ * V_SZ;
    float* out_h    = out_ctx  + B_SZ * N_SZ;
    float* out_attn = out_h    + B_SZ * N_SZ;

    k_embed_concat<<<B_SZ, 256, 0, stream>>>(word, last_ctx, emb, rnn);

    // GRU gate GEMMs (fp32 WMMA): gx = rnn @ W_ih^T + b_ih ; gh = h @ W_hh^T + b_hh
    k_wmma_gemm_awt<<<(G3N / 16 + 7) / 8, 256, 0, stream>>>(rnn, W_ih, b_ih, gx, 2 * N_SZ, G3N);
    k_wmma_gemm_awt<<<(G3N / 16 + 7) / 8, 256, 0, stream>>>(last_h, W_hh, b_hh, gh, N_SZ, G3N);

    k_gru_gates<<<(B_SZ * N_SZ) / 256, 256, 0, stream>>>(gx, gh, last_h, hnew, cat, out_h);

    // single-pass chunked attention over the 512MB encoder tensor
    dim3 ga(NCHUNK, B_SZ);
    k_attn_partial<<<ga, 256, 0, stream>>>(hnew, enc, out_attn, pctx, lmax, lsum);
    k_attn_combine<<<B_SZ, 256, 0, stream>>>(lmax, lsum, pctx, out_attn, out_ctx, cat);

    // logits = [h_new | context] @ W_out^T + b_out  (412MB weight stream, WMMA fp32)
    const int ntiles = (V_SZ + 15) / 16;
    k_wmma_gemm_awt<<<(ntiles + 7) / 8, 256, 0, stream>>>(cat, W_out, b_out, logits, 2 * N_SZ, V_SZ);

    k_log_softmax<<<B_SZ, 256, 0, stream>>>(logits, out_logp);
}